// SPDLayer_56616258896178
// MI455X (gfx1250) — compile-verified
//
#include <hip/hip_runtime.h>
#include <math.h>

// SPD layer on MI455X (gfx1250): eigh-free formulation.
//   X_tan = logm(eps*I + relu_spec(X))          (128x128, per batch)
//   G     = X_tan[:64,:] @ W[:64,:]^T           (64x64)
//   out   = spectral_clip_eps(sym_lower(I + G)) (64x64)
// All spectral ops via Newton-Schulz sign / inverse scaling-and-squaring,
// i.e. pure matmuls -> v_wmma_f32_16x16x4_f32.

#define NTHREADS 256
#define NWAVES   8
#define N1       128
#define N2       64
#define EPSV     1e-3f

typedef __attribute__((ext_vector_type(2))) float v2f;
typedef __attribute__((ext_vector_type(8))) float v8f;

// Cooperative NxN matmul D = A*B, row-major in LDS, 8 waves per block.
// WMMA f32 16x16x4 layouts (wave32):
//   A tile  : lane l, vgpr v -> (M = l%16, K = 2*(l/16)+v)  => one 8B load/lane/k-step
//   B tile  : lane l, vgpr v -> (K = 2*(l/16)+v, N = l%16)
//   C/D tile: lane l, vgpr r -> (M = r + 8*(l/16), N = l%16)
template <int N>
__device__ __forceinline__ void mm_lds(float* D, const float* A, const float* B,
                                       int wave, int lane) {
  constexpr int NT = N / 16;
  const int lr = lane & 15;
  const int lh = lane >> 4;
  for (int t = wave; t < NT * NT; t += NWAVES) {
    const int tm = t / NT;
    const int tn = t % NT;
    v8f acc = {};
    const float* arow = A + (tm * 16 + lr) * N + 2 * lh;
    const float* bcol = B + (2 * lh) * N + tn * 16 + lr;
    for (int k = 0; k < N; k += 4) {
      v2f a = *(const v2f*)(arow + k);          // ds_load_b64
      v2f bv;
      bv.x = bcol[k * N];
      bv.y = bcol[k * N + N];
      acc = __builtin_amdgcn_wmma_f32_16x16x4_f32(
          false, a, false, bv, (short)0, acc, false, false);
    }
    float* drow = D + (tm * 16 + 8 * lh) * N + tn * 16 + lr;
#pragma unroll
    for (int r = 0; r < 8; ++r) drow[r * N] = acc[r];
  }
}

__device__ __forceinline__ float block_frob(const float* A, int n, int tid, float* red) {
  float s = 0.f;
  for (int i = tid; i < n; i += NTHREADS) { float v = A[i]; s += v * v; }
  red[tid] = s;
  __syncthreads();
  for (int off = NTHREADS / 2; off > 0; off >>= 1) {
    if (tid < off) red[tid] += red[tid + off];
    __syncthreads();
  }
  float r = sqrtf(red[0]);
  __syncthreads();
  return r;
}

__global__ void __launch_bounds__(NTHREADS)
spd_layer_kernel(const float* __restrict__ X, const float* __restrict__ Wg,
                 float* __restrict__ out) {
  extern __shared__ float lds[];          // 4 x 128x128 f32 = 256 KB
  float* B0 = lds;
  float* B1 = lds + 16384;
  float* B2 = lds + 2 * 16384;
  float* B3 = lds + 3 * 16384;
  __shared__ float red[NTHREADS];

  const int tid  = threadIdx.x;
  const int wave = tid >> 5;
  const int lane = tid & 31;
  const int b    = blockIdx.x;
  const float* Xb = X + (size_t)b * (N1 * N1);

  // ---------- Stage 1a: spectral relu of X (note Xp - eps*I == X) ----------
  float nrm = block_frob(Xb, N1 * N1, tid, red) + 1e-20f;
  float inv = 1.f / nrm;
  for (int i = tid; i < N1 * N1; i += NTHREADS) B0[i] = Xb[i] * inv;   // Z0
  __syncthreads();

  for (int it = 0; it < 14; ++it) {       // Newton-Schulz sign(X)
    mm_lds<N1>(B1, B0, B0, wave, lane); __syncthreads();               // T = Z*Z
    mm_lds<N1>(B2, B0, B1, wave, lane); __syncthreads();               // U = Z*T
    for (int i = tid; i < N1 * N1; i += NTHREADS)
      B0[i] = 1.5f * B0[i] - 0.5f * B2[i];
    __syncthreads();
  }
  for (int i = tid; i < N1 * N1; i += NTHREADS) B1[i] = Xb[i];
  __syncthreads();
  mm_lds<N1>(B2, B1, B0, wave, lane); __syncthreads();                 // X*sign(X)
  for (int i = tid; i < N1 * N1; i += NTHREADS) {                      // A = eps*I + relu(X)
    int r = i >> 7, c = i & 127;
    B0[i] = 0.5f * (B1[i] + B2[i]) + ((r == c) ? EPSV : 0.f);
  }
  __syncthreads();

  // ---------- Stage 1b: log(A) by inverse scaling-and-squaring ----------
  float cnorm = block_frob(B0, N1 * N1, tid, red) + 1e-20f;            // spectrum(A/c) in (0,1]
  float invc = 1.f / cnorm;
  for (int i = tid; i < N1 * N1; i += NTHREADS) B0[i] *= invc;
  __syncthreads();

  float *Y = B0, *Zi = B1, *Tb = B2, *Sp = B3;
  for (int root = 0; root < 5; ++root) {                               // 5 coupled NS sqrts
    for (int i = tid; i < N1 * N1; i += NTHREADS) {
      int r = i >> 7, c = i & 127;
      Zi[i] = (r == c) ? 1.f : 0.f;
    }
    __syncthreads();
    for (int it = 0; it < 7; ++it) {
      mm_lds<N1>(Tb, Zi, Y, wave, lane); __syncthreads();              // T = Z*Y
      for (int i = tid; i < N1 * N1; i += NTHREADS) {
        int r = i >> 7, c = i & 127;
        Tb[i] = ((r == c) ? 1.5f : 0.f) - 0.5f * Tb[i];                // T = (3I - ZY)/2
      }
      __syncthreads();
      mm_lds<N1>(Sp, Y, Tb, wave, lane); __syncthreads();              // newY = Y*T
      mm_lds<N1>(Y,  Tb, Zi, wave, lane); __syncthreads();             // newZ = T*Z (into old Y)
      float* t0 = Y;  Y = Sp;  Sp = Zi;  Zi = t0;                      // rotate buffers
    }
  }
  // Mercator series: log(R) ~ E - E^2/2 + E^3/3 - E^4/4, E = R - I
  float* E = Zi;
  for (int i = tid; i < N1 * N1; i += NTHREADS) {
    int r = i >> 7, c = i & 127;
    E[i] = Y[i] - ((r == c) ? 1.f : 0.f);
  }
  __syncthreads();
  float* E2 = Tb; mm_lds<N1>(E2, E,  E,  wave, lane); __syncthreads();
  float* E3 = Sp; mm_lds<N1>(E3, E2, E,  wave, lane); __syncthreads();
  float* E4 = Y;  mm_lds<N1>(E4, E2, E2, wave, lane); __syncthreads();
  const float logc = logf(cnorm);
  for (int i = tid; i < N1 * N1; i += NTHREADS) {                      // X_tan (in place of E)
    int r = i >> 7, c = i & 127;
    float p = E[i] - 0.5f * E2[i] + (1.f / 3.f) * E3[i] - 0.25f * E4[i];
    E[i] = 32.f * p + ((r == c) ? logc : 0.f);
  }
  __syncthreads();
  float* Xt = E;            // Zi buffer
  float* Gm = E2;           // Tb buffer, will hold 64x64 G

  // ---------- Stage 2: G = X_tan[:64,:] * W^T  (B operand streamed from HBM) ----------
  {
    const int lr = lane & 15, lh = lane >> 4;
    for (int t = wave; t < 16; t += NWAVES) {
      const int tm = t >> 2, tn = t & 3;
      v8f acc = {};
      const float* arow = Xt + (tm * 16 + lr) * N1 + 2 * lh;
      const float* wrow = Wg + (tn * 16 + lr) * N1 + 2 * lh;           // W[o][k], contiguous k pair
      for (int k = 0; k < N1; k += 4) {
        v2f a  = *(const v2f*)(arow + k);
        v2f bv = *(const v2f*)(wrow + k);                              // global_load_b64
        acc = __builtin_amdgcn_wmma_f32_16x16x4_f32(
            false, a, false, bv, (short)0, acc, false, false);
      }
      float* drow = Gm + (tm * 16 + 8 * lh) * N2 + tn * 16 + lr;
#pragma unroll
      for (int r = 0; r < 8; ++r) drow[r * N2] = acc[r];
    }
    __syncthreads();
  }

  // ---------- Stage 3: spectral clip of sym_lower(I + G) ----------
  float* Ms = E3;                                                      // Sp buffer
  for (int i = tid; i < N2 * N2; i += NTHREADS) {
    int r = i >> 6, c = i & 63;
    float g = (r >= c) ? Gm[r * N2 + c] : Gm[c * N2 + r];              // eigh uses lower tri
    Ms[i] = g + ((r == c) ? (1.f - EPSV) : 0.f);                       // M = S - eps*I
  }
  __syncthreads();
  float n2 = block_frob(Ms, N2 * N2, tid, red) + 1e-20f;
  float* Zs  = E4;                                                     // Y buffer
  float inv2 = 1.f / n2;
  for (int i = tid; i < N2 * N2; i += NTHREADS) Zs[i] = Ms[i] * inv2;
  __syncthreads();
  float* T64 = Xt;                                                     // Zi buffer
  float* U64 = Gm;                                                     // Tb buffer
  for (int it = 0; it < 14; ++it) {
    mm_lds<N2>(T64, Zs, Zs,  wave, lane); __syncthreads();
    mm_lds<N2>(U64, Zs, T64, wave, lane); __syncthreads();
    for (int i = tid; i < N2 * N2; i += NTHREADS)
      Zs[i] = 1.5f * Zs[i] - 0.5f * U64[i];
    __syncthreads();
  }
  mm_lds<N2>(U64, Ms, Zs, wave, lane); __syncthreads();                // M*sign(M)
  float* ob = out + (size_t)b * (N2 * N2);
  for (int i = tid; i < N2 * N2; i += NTHREADS) {
    int r = i >> 6, c = i & 63;
    ob[i] = 0.5f * (Ms[i] + U64[i]) + ((r == c) ? EPSV : 0.f);         // eps*I + relu(M)
  }
}

extern "C" void kernel_launch(void* const* d_in, const int* in_sizes, int n_in,
                              void* d_out, int out_size, void* d_ws, size_t ws_size,
                              hipStream_t stream) {
  (void)in_sizes; (void)n_in; (void)out_size; (void)d_ws; (void)ws_size;
  const float* X = (const float*)d_in[0];   // (2048,128,128) f32
  const float* W = (const float*)d_in[1];   // (64,128) f32
  float* out = (float*)d_out;               // (2048,64,64) f32

  const size_t lds_bytes = 4u * N1 * N1 * sizeof(float);  // 256 KB dynamic LDS
  hipFuncSetAttribute((const void*)spd_layer_kernel,
                      hipFuncAttributeMaxDynamicSharedMemorySize, (int)lds_bytes);
  spd_layer_kernel<<<2048, NTHREADS, lds_bytes, stream>>>(X, W, out);
}